// Stage_GNN_learner_9500467658952
// MI455X (gfx1250) — compile-verified
//
#include <hip/hip_runtime.h>
#include <hip/hip_bf16.h>
#include <math.h>

// ---------------------------------------------------------------------------
// Stage GNN learner, MI455X (gfx1250), f32 WMMA path (V_WMMA_F32_16X16X4_F32)
// N=4000, ISIZE=512, OSIZE=256, VPOS=64, KS=(0.5,0.5), K=30, EPS=0.25
// Forward-pass simplification: the score mask multiply is identity in forward,
// so emb == embeddings at every stage and vpos_map(down_outs[j]) == VM for all j.
// ---------------------------------------------------------------------------

#define NN_N   4000
#define IN_SZ  512
#define OS_SZ  256
#define VP_SZ  64
#define CAT_SZ (OS_SZ + VP_SZ)   // 320
#define KNN_K  31
#define EPS_V  0.25f

typedef __attribute__((ext_vector_type(2))) float v2f;
typedef __attribute__((ext_vector_type(8))) float v8f;

// ============================ WMMA f32 GEMM ================================
// C[M x Nc] = act(A[M x K] @ B (+bias)) (+resid).  One wave per 16x32 strip.
// Requires: M % 16 == 0, Nc % 32 == 0, K % 4 == 0 (true for every call here).
// Fragment layouts per CDNA5 ISA 7.12.2 (32-bit A 16x4 / B 4x16 / C 16x16).
template <bool TRANSB, bool RELU, bool BIAS, bool RESID>
__global__ __launch_bounds__(32) void wmma_gemm(
    const float* __restrict__ A, int lda,
    const float* __restrict__ B, int ldb,
    float* __restrict__ C, int ldc, int K,
    const float* __restrict__ bias,
    const float* __restrict__ resid, int ldr) {
  const int lane = threadIdx.x;
  const int l15  = lane & 15;
  const int half = lane >> 4;       // 0: K={0,1}, 1: K={2,3} within a step
  const int m0   = blockIdx.y << 4;
  const int n0   = blockIdx.x << 5; // two 16-wide output tiles
  const int kA   = half << 1;

  v8f acc0 = {0.f,0.f,0.f,0.f,0.f,0.f,0.f,0.f};
  v8f acc1 = {0.f,0.f,0.f,0.f,0.f,0.f,0.f,0.f};

  const float* Arow = A + (size_t)(m0 + l15) * lda + kA;
  for (int k = 0; k < K; k += 4) {
    v2f a, b0, b1;
    a[0] = Arow[k];
    a[1] = Arow[k + 1];
    if (TRANSB) {
      // B is stored row-major [Nc x K]; logical B[k][n] = Bt[n][k]
      const float* B0 = B + (size_t)(n0 + l15) * ldb + k + kA;
      const float* B1 = B + (size_t)(n0 + 16 + l15) * ldb + k + kA;
      b0[0] = B0[0]; b0[1] = B0[1];
      b1[0] = B1[0]; b1[1] = B1[1];
    } else {
      const float* Bk = B + (size_t)(k + kA) * ldb + n0 + l15;
      b0[0] = Bk[0];   b0[1] = Bk[ldb];
      b1[0] = Bk[16];  b1[1] = Bk[ldb + 16];
    }
    acc0 = __builtin_amdgcn_wmma_f32_16x16x4_f32(false, a, false, b0,
                                                 (short)0, acc0, false, false);
    acc1 = __builtin_amdgcn_wmma_f32_16x16x4_f32(false, a, false, b1,
                                                 (short)0, acc1, false, false);
  }
#pragma unroll
  for (int v = 0; v < 8; ++v) {
    int row = m0 + v + (half ? 8 : 0);
    int c0 = n0 + l15, c1 = n0 + 16 + l15;
    float v0 = acc0[v], v1 = acc1[v];
    if (BIAS)  { v0 += bias[c0]; v1 += bias[c1]; }
    if (RELU)  { v0 = fmaxf(v0, 0.f); v1 = fmaxf(v1, 0.f); }
    if (RESID) { v0 += resid[(size_t)row * ldr + c0];
                 v1 += resid[(size_t)row * ldr + c1]; }
    C[(size_t)row * ldc + c0] = v0;
    C[(size_t)row * ldc + c1] = v1;
  }
}

// ============================ small kernels ================================

__global__ __launch_bounds__(256) void rownorm_k(const float* __restrict__ X,
                                                 float* __restrict__ Y) {
  __shared__ float red[256];
  const int r = blockIdx.x, t = threadIdx.x;
  float v = X[(size_t)r * OS_SZ + t];
  red[t] = v * v; __syncthreads();
  for (int s = 128; s > 0; s >>= 1) { if (t < s) red[t] += red[t + s]; __syncthreads(); }
  float inv = 1.0f / fmaxf(sqrtf(red[0]), 1e-8f);
  Y[(size_t)r * OS_SZ + t] = v * inv;
}

__global__ __launch_bounds__(256) void rowsum_k(const float* __restrict__ A,
                                                float* __restrict__ d, int w) {
  __shared__ float red[256];
  const int r = blockIdx.x, t = threadIdx.x;
  float s = 0.f;
  for (int c = t; c < w; c += 256) s += A[(size_t)r * w + c];
  red[t] = s; __syncthreads();
  for (int st = 128; st > 0; st >>= 1) { if (t < st) red[t] += red[t + st]; __syncthreads(); }
  if (t == 0) d[r] = red[0];
}

__global__ void colsum_k(const float* __restrict__ A, float* __restrict__ d, int n) {
  int c = blockIdx.x * blockDim.x + threadIdx.x;
  if (c >= n) return;
  float s = 0.f;
  for (int r = 0; r < n; ++r) s += A[(size_t)r * n + c];
  d[c] = s;
}

// dis[i] = d>0 ? d^-0.5 : 0, where d = scale*(a[i] + (b?b[i]:0))
__global__ void dis_k(const float* __restrict__ a, const float* __restrict__ b,
                      float* __restrict__ dis, int n, float scale) {
  int i = blockIdx.x * blockDim.x + threadIdx.x;
  if (i >= n) return;
  float d = a[i] + (b ? b[i] : 0.f);
  d *= scale;
  dis[i] = (d > 0.f) ? (1.0f / sqrtf(d)) : 0.f;
}

// out = 0.5*(L + L^T) * dis_r * dis_c   (tiled transpose, 32x32 tiles)
__global__ __launch_bounds__(256) void la_tiled_k(const float* __restrict__ L,
                                                  const float* __restrict__ dis,
                                                  float* __restrict__ out, int n) {
  __shared__ float t0[32][33];
  __shared__ float t1[32][33];
  const int bx = blockIdx.x * 32, by = blockIdx.y * 32;
  for (int i = threadIdx.y; i < 32; i += 8) {
    t0[i][threadIdx.x] = L[(size_t)(by + i) * n + bx + threadIdx.x];
    t1[i][threadIdx.x] = L[(size_t)(bx + i) * n + by + threadIdx.x];
  }
  __syncthreads();
  for (int i = threadIdx.y; i < 32; i += 8) {
    int r = by + i, c = bx + threadIdx.x;
    float v = 0.5f * (t0[i][threadIdx.x] + t1[threadIdx.x][i]);
    out[(size_t)r * n + c] = v * dis[r] * dis[c];
  }
}

// out = A * dis_r * dis_c (A symmetric already)
__global__ void scale_sym_k(const float* __restrict__ A, const float* __restrict__ dis,
                            float* __restrict__ out, int n) {
  int t = blockIdx.x * blockDim.x + threadIdx.x;
  if (t >= n * n) return;
  int r = t / n, c = t - r * n;
  out[t] = A[t] * dis[r] * dis[c];
}

__global__ __launch_bounds__(256) void gemv256_k(const float* __restrict__ A,
                                                 const float* __restrict__ w,
                                                 float* __restrict__ out) {
  __shared__ float red[256];
  const int r = blockIdx.x, t = threadIdx.x;
  red[t] = A[(size_t)r * OS_SZ + t] * w[t]; __syncthreads();
  for (int s = 128; s > 0; s >>= 1) { if (t < s) red[t] += red[t + s]; __syncthreads(); }
  if (t == 0) out[r] = red[0];
}

__global__ __launch_bounds__(256) void matvec_sig_k(const float* __restrict__ A,
                                                    const float* __restrict__ x,
                                                    float* __restrict__ y, int n) {
  __shared__ float red[256];
  const int r = blockIdx.x, t = threadIdx.x;
  float s = 0.f;
  for (int c = t; c < n; c += 256) s += A[(size_t)r * n + c] * x[c];
  red[t] = s; __syncthreads();
  for (int st = 128; st > 0; st >>= 1) { if (t < st) red[t] += red[t + st]; __syncthreads(); }
  if (t == 0) y[r] = 1.0f / (1.0f + expf(-red[0]));
}

__global__ __launch_bounds__(256) void gather_rowsum_k(const float* __restrict__ ADJ,
                                                       const int* __restrict__ idx,
                                                       int m, float* __restrict__ d) {
  __shared__ float red[256];
  const int r = blockIdx.x, t = threadIdx.x;
  const size_t base = (size_t)idx[r] * NN_N;
  float s = 0.f;
  for (int c = t; c < m; c += 256) s += ADJ[base + idx[c]];
  red[t] = s; __syncthreads();
  for (int st = 128; st > 0; st >>= 1) { if (t < st) red[t] += red[t + st]; __syncthreads(); }
  if (t == 0) d[r] = red[0];
}

__global__ __launch_bounds__(256) void gather_gemv256_k(const float* __restrict__ A,
                                                        const int* __restrict__ idx,
                                                        const float* __restrict__ w,
                                                        float* __restrict__ out) {
  __shared__ float red[256];
  const int r = blockIdx.x, t = threadIdx.x;
  red[t] = A[(size_t)idx[r] * OS_SZ + t] * w[t]; __syncthreads();
  for (int s = 128; s > 0; s >>= 1) { if (t < s) red[t] += red[t + s]; __syncthreads(); }
  if (t == 0) out[r] = red[0];
}

// y[r] = sigmoid(dis[r] * sum_c adj[p r, p c] * dis[c] * s2[c])
__global__ __launch_bounds__(256) void gather_norm_sig_k(const float* __restrict__ ADJ,
                                                         const int* __restrict__ idx, int m,
                                                         const float* __restrict__ dis,
                                                         const float* __restrict__ s2,
                                                         float* __restrict__ y) {
  __shared__ float red[256];
  const int r = blockIdx.x, t = threadIdx.x;
  const size_t base = (size_t)idx[r] * NN_N;
  float s = 0.f;
  for (int c = t; c < m; c += 256) s += ADJ[base + idx[c]] * dis[c] * s2[c];
  red[t] = s; __syncthreads();
  for (int st = 128; st > 0; st >>= 1) { if (t < st) red[t] += red[t + st]; __syncthreads(); }
  if (t == 0) y[r] = 1.0f / (1.0f + expf(-dis[r] * red[0]));
}

// rank[i] = #{j : y[j] > y[i]  or (y[j]==y[i] and j<i)}   (jax top_k tie-break)
__global__ void rank_k(const float* __restrict__ y, int n, int* __restrict__ rank) {
  int i = blockIdx.x * blockDim.x + threadIdx.x;
  if (i >= n) return;
  float yi = y[i];
  int r = 0;
  for (int j = 0; j < n; ++j) {
    float yj = y[j];
    r += (yj > yi) || (yj == yi && j < i);
  }
  rank[i] = r;
}

// Write indices with rank<k, in ascending index order (== top_k + argsort(idx)).
__global__ void compact_k(const int* __restrict__ rank, int n, int k,
                          const int* __restrict__ prev, int* __restrict__ out) {
  if (threadIdx.x != 0 || blockIdx.x != 0) return;
  int c = 0;
  for (int i = 0; i < n; ++i)
    if (rank[i] < k) out[c++] = prev ? prev[i] : i;
}

__global__ void fill_i32_k(int* p, int n, int v) {
  int i = blockIdx.x * blockDim.x + threadIdx.x;
  if (i < n) p[i] = v;
}
__global__ void scatter_i32_k(int* p, const int* idx, int m, int v) {
  int i = blockIdx.x * blockDim.x + threadIdx.x;
  if (i < m) p[idx[i]] = v;
}

__global__ void concat_vpos_k(const float* __restrict__ emb, const float* __restrict__ evp,
                              const int* __restrict__ vp, float* __restrict__ e2, int n) {
  int t = blockIdx.x * blockDim.x + threadIdx.x;
  if (t >= n * CAT_SZ) return;
  int r = t / CAT_SZ, c = t - r * CAT_SZ;
  e2[t] = (c < OS_SZ) ? emb[(size_t)r * OS_SZ + c] : evp[vp[r] * VP_SZ + (c - OS_SZ)];
}

__global__ void add_vec_k(const float* a, const float* b, float* o, int n) {
  int i = blockIdx.x * blockDim.x + threadIdx.x;
  if (i < n) o[i] = a[i] + b[i];
}

__global__ void blend_k(float* __restrict__ adjc, const float* __restrict__ la,
                        const int* __restrict__ idx, int m, int n) {
  int t = blockIdx.x * blockDim.x + threadIdx.x;
  if (t >= m * m) return;
  int r = t / m, c = t - r * m;
  size_t o = (size_t)idx[r] * n + idx[c];
  adjc[o] = 0.5f * la[o] + 0.5f * adjc[o];
}

__global__ void pred_k(const float* __restrict__ la, const float* __restrict__ adjc,
                       float* __restrict__ pred, int total) {
  int t = blockIdx.x * blockDim.x + threadIdx.x;
  if (t < total) pred[t] = 0.5f * la[t] + 0.5f * adjc[t];
}

// ============== per-row radix-select top-k mask + epsilon, in place ========
__device__ __forceinline__ unsigned sortkey(float f) {
  unsigned u = __float_as_uint(f);
  return (u & 0x80000000u) ? ~u : (u | 0x80000000u);
}

__global__ __launch_bounds__(256) void knn_enn_k(float* __restrict__ M, int n,
                                                 int k, float eps) {
  __shared__ int redi[256];
  __shared__ int wcnt[8];
  __shared__ int tieBaseS;
  const int tid = threadIdx.x;
  float* R = M + (size_t)blockIdx.x * n;

  // radix select: T = k-th largest sortable key in the row
  unsigned prefix = 0;
  int remaining = k;
  for (int bit = 31; bit >= 0; --bit) {
    const unsigned cand = prefix | (1u << bit);
    const unsigned hi = ~((1u << bit) - 1u);
    int cnt = 0;
    for (int c = tid; c < n; c += 256) cnt += ((sortkey(R[c]) & hi) == cand);
    redi[tid] = cnt; __syncthreads();
    for (int st = 128; st > 0; st >>= 1) { if (tid < st) redi[tid] += redi[tid + st]; __syncthreads(); }
    int total = redi[0];
    __syncthreads();
    if (total >= remaining) prefix = cand;
    else remaining -= total;
  }
  const unsigned T = prefix;

  int cg = 0;
  for (int c = tid; c < n; c += 256) cg += (sortkey(R[c]) > T);
  redi[tid] = cg; __syncthreads();
  for (int st = 128; st > 0; st >>= 1) { if (tid < st) redi[tid] += redi[tid + st]; __syncthreads(); }
  const int remTies = k - redi[0];   // ties at T to accept (lowest column index first)
  __syncthreads();
  if (tid == 0) tieBaseS = 0;
  __syncthreads();

  const int lane = tid & 31, wv = tid >> 5;
  for (int base = 0; base < n; base += 256) {
    int c = base + tid;
    float f = (c < n) ? R[c] : -3.0e38f;
    unsigned s = sortkey(f);
    int eq = (c < n) && (s == T);
    unsigned long long bal = __ballot(eq);
    int lanePre = __popcll(bal & ((1ull << lane) - 1ull));
    if (lane == 0) wcnt[wv] = __popcll(bal);
    __syncthreads();
    int woff = 0, totalEq = 0;
    for (int w = 0; w < 8; ++w) { if (w < wv) woff += wcnt[w]; totalEq += wcnt[w]; }
    bool sel = (s > T) || (eq && (tieBaseS + woff + lanePre) < remTies);
    if (c < n) R[c] = (sel && f > eps) ? f : 0.0f;
    __syncthreads();
    if (tid == 0) tieBaseS += totalEq;
    __syncthreads();
  }
}

// ================================ driver ===================================
extern "C" void kernel_launch(void* const* d_in, const int* in_sizes, int n_in,
                              void* d_out, int out_size, void* d_ws, size_t ws_size,
                              hipStream_t stream) {
  (void)in_sizes; (void)n_in; (void)out_size; (void)ws_size;

  const float* F   = (const float*)d_in[0];   // 4000x512
  const float* ADJ = (const float*)d_in[1];   // 4000x4000
  const float* W11 = (const float*)d_in[2];   // 512x256
  const float* W12 = (const float*)d_in[3];   // 256x256
  const float* WSC = (const float*)d_in[4];   // 256x1
  const float* W21 = (const float*)d_in[5];   // 256x256
  const float* W22 = (const float*)d_in[6];   // 256x256
  const float* EVP = (const float*)d_in[7];   // 3x64
  const float* WL1 = (const float*)d_in[8];   // 320x256
  const float* BL1 = (const float*)d_in[9];   // 256
  const float* WL2 = (const float*)d_in[10];  // 256x256
  const float* BL2 = (const float*)d_in[11];  // 256

  const size_t NN = (size_t)NN_N * NN_N;
  float* W    = (float*)d_ws;
  float* L    = W;                  // cosine / learned_adj (in-place knn+enn)
  float* LA   = W + NN;             // la, then reused as norm(adj_cur)
  float* ADJC = W + 2 * NN;         // adj_cur
  float* EMB  = W + 3 * NN;                        // 4000x256
  float* VM   = EMB + (size_t)NN_N * OS_SZ;        // 4000x256
  float* TA   = VM  + (size_t)NN_N * OS_SZ;        // 4000x256 scratch
  float* TB   = TA  + (size_t)NN_N * OS_SZ;        // 4000x256 scratch
  float* E2   = TB  + (size_t)NN_N * OS_SZ;        // 4000x320
  float* XN   = E2  + (size_t)NN_N * CAT_SZ;       // 4000x256
  float* V0   = XN  + (size_t)NN_N * OS_SZ;        // vectors (4096 each)
  float* V1   = V0 + 4096;
  float* DIS  = V1 + 4096;
  float* SV   = DIS + 4096;
  float* YV   = SV + 4096;
  float* S2   = YV + 4096;
  float* DIS2 = S2 + 4096;
  float* Y2   = DIS2 + 4096;
  int* RANKS  = (int*)(Y2 + 4096);
  int* IDX0   = RANKS + 4096;       // 2000 sorted indices (stage 0)
  int* IDX1   = IDX0 + 4096;        // 1000 sorted indices (stage 1)
  int* VPI    = IDX1 + 4096;        // vp per node

  float* OUT_LA   = (float*)d_out;
  float* OUT_PRED = OUT_LA + NN;

  const dim3 g256(OS_SZ / 32, NN_N / 16);   // (8,250) for 4000x256 outputs
  const dim3 gNN(NN_N / 32, NN_N / 16);     // (125,250) for 4000x4000 outputs
  const dim3 gla(NN_N / 32, NN_N / 32);     // (125,125) tiled sym kernels
  const dim3 b32(32), b256(256), bt(32, 8);

  // ---- encoder-1 GCN: embeddings = adj @ relu(adj @ (F @ W11)) @ W12 ------
  wmma_gemm<false,false,false,false><<<g256, b32, 0, stream>>>(F,   IN_SZ, W11, OS_SZ, TA,  OS_SZ, IN_SZ, nullptr, nullptr, 0);
  wmma_gemm<false,true, false,false><<<g256, b32, 0, stream>>>(ADJ, NN_N,  TA,  OS_SZ, TB,  OS_SZ, NN_N,  nullptr, nullptr, 0);
  wmma_gemm<false,false,false,false><<<g256, b32, 0, stream>>>(TB,  OS_SZ, W12, OS_SZ, TA,  OS_SZ, OS_SZ, nullptr, nullptr, 0);
  wmma_gemm<false,false,false,false><<<g256, b32, 0, stream>>>(ADJ, NN_N,  TA,  OS_SZ, EMB, OS_SZ, NN_N,  nullptr, nullptr, 0);

  // ---- pooling stage 0: y = sigmoid(adj @ (emb @ w_score)); top 2000 ------
  gemv256_k<<<NN_N, b256, 0, stream>>>(EMB, WSC, SV);
  matvec_sig_k<<<NN_N, b256, 0, stream>>>(ADJ, SV, YV, NN_N);
  rank_k<<<(NN_N + 255) / 256, b256, 0, stream>>>(YV, NN_N, RANKS);
  compact_k<<<1, b32, 0, stream>>>(RANKS, NN_N, 2000, nullptr, IDX0);

  // ---- pooling stage 1 on 2000x2000 sym-normed submatrix; top 1000 --------
  gather_rowsum_k<<<2000, b256, 0, stream>>>(ADJ, IDX0, 2000, V0);
  dis_k<<<8, b256, 0, stream>>>(V0, nullptr, DIS2, 2000, 1.0f);
  gather_gemv256_k<<<2000, b256, 0, stream>>>(EMB, IDX0, WSC, S2);
  gather_norm_sig_k<<<2000, b256, 0, stream>>>(ADJ, IDX0, 2000, DIS2, S2, Y2);
  rank_k<<<(2000 + 255) / 256, b256, 0, stream>>>(Y2, 2000, RANKS);
  compact_k<<<1, b32, 0, stream>>>(RANKS, 2000, 1000, IDX0, IDX1);

  // ---- vp embedding + vpos_map (VM shared across all stages) --------------
  fill_i32_k<<<(NN_N + 255) / 256, b256, 0, stream>>>(VPI, NN_N, 0);
  scatter_i32_k<<<8, b256, 0, stream>>>(VPI, IDX0, 2000, 1);
  scatter_i32_k<<<4, b256, 0, stream>>>(VPI, IDX1, 1000, 2);
  concat_vpos_k<<<(NN_N * CAT_SZ + 255) / 256, b256, 0, stream>>>(EMB, EVP, VPI, E2, NN_N);
  wmma_gemm<false,true, true, false><<<g256, b32, 0, stream>>>(E2, CAT_SZ, WL1, OS_SZ, TA, OS_SZ, CAT_SZ, BL1, nullptr, 0);
  wmma_gemm<false,false,true, false><<<g256, b32, 0, stream>>>(TA, OS_SZ,  WL2, OS_SZ, VM, OS_SZ, OS_SZ,  BL2, nullptr, 0);

  // ---- first learned adjacency: cosine(VM + embeddings), knn+enn ----------
  add_vec_k<<<(NN_N * OS_SZ + 255) / 256, b256, 0, stream>>>(VM, EMB, TB, NN_N * OS_SZ);
  rownorm_k<<<NN_N, b256, 0, stream>>>(TB, XN);
  wmma_gemm<true, false,false,false><<<gNN, b32, 0, stream>>>(XN, OS_SZ, XN, OS_SZ, L, NN_N, OS_SZ, nullptr, nullptr, 0);
  knn_enn_k<<<NN_N, b256, 0, stream>>>(L, NN_N, KNN_K, EPS_V);

  hipMemcpyAsync(ADJC, ADJ, NN * sizeof(float), hipMemcpyDeviceToDevice, stream);

  // ---- unpool loop: j = 1 then j = 0 --------------------------------------
  for (int jj = 0; jj < 2; ++jj) {
    const int* idx = (jj == 0) ? IDX1 : IDX0;
    const int  m   = (jj == 0) ? 1000 : 2000;

    // la = sym_norm(0.5*(L + L^T))
    rowsum_k<<<NN_N, b256, 0, stream>>>(L, V0, NN_N);
    colsum_k<<<(NN_N + 255) / 256, b256, 0, stream>>>(L, V1, NN_N);
    dis_k<<<(NN_N + 255) / 256, b256, 0, stream>>>(V0, V1, DIS, NN_N, 0.5f);
    la_tiled_k<<<gla, bt, 0, stream>>>(L, DIS, LA, NN_N);

    // blend submatrix of adj_cur with la
    blend_k<<<(m * m + 255) / 256, b256, 0, stream>>>(ADJC, LA, idx, m, NN_N);

    // norm_adj = sym_norm(adj_cur) (symmetric) -> reuse LA buffer
    rowsum_k<<<NN_N, b256, 0, stream>>>(ADJC, V0, NN_N);
    dis_k<<<(NN_N + 255) / 256, b256, 0, stream>>>(V0, nullptr, DIS, NN_N, 1.0f);
    scale_sym_k<<<(int)((NN + 255) / 256), b256, 0, stream>>>(ADJC, DIS, LA, NN_N);

    // e = gcn2(VM, norm_adj, W21, W22) + embeddings
    wmma_gemm<false,false,false,false><<<g256, b32, 0, stream>>>(VM, OS_SZ, W21, OS_SZ, TA, OS_SZ, OS_SZ, nullptr, nullptr, 0);
    wmma_gemm<false,true, false,false><<<g256, b32, 0, stream>>>(LA, NN_N,  TA,  OS_SZ, TB, OS_SZ, NN_N,  nullptr, nullptr, 0);
    wmma_gemm<false,false,false,false><<<g256, b32, 0, stream>>>(TB, OS_SZ, W22, OS_SZ, TA, OS_SZ, OS_SZ, nullptr, nullptr, 0);
    wmma_gemm<false,false,false,true ><<<g256, b32, 0, stream>>>(LA, NN_N,  TA,  OS_SZ, TB, OS_SZ, NN_N,  nullptr, EMB, OS_SZ);

    // learned_adj = enn(knn(cosine(e)))
    rownorm_k<<<NN_N, b256, 0, stream>>>(TB, XN);
    wmma_gemm<true, false,false,false><<<gNN, b32, 0, stream>>>(XN, OS_SZ, XN, OS_SZ, L, NN_N, OS_SZ, nullptr, nullptr, 0);
    knn_enn_k<<<NN_N, b256, 0, stream>>>(L, NN_N, KNN_K, EPS_V);
  }

  // ---- outputs: la (first half of d_out), pred = 0.5*la + 0.5*adj_cur -----
  rowsum_k<<<NN_N, b256, 0, stream>>>(L, V0, NN_N);
  colsum_k<<<(NN_N + 255) / 256, b256, 0, stream>>>(L, V1, NN_N);
  dis_k<<<(NN_N + 255) / 256, b256, 0, stream>>>(V0, V1, DIS, NN_N, 0.5f);
  la_tiled_k<<<gla, bt, 0, stream>>>(L, DIS, OUT_LA, NN_N);
  pred_k<<<(int)((NN + 255) / 256), b256, 0, stream>>>(OUT_LA, ADJC, OUT_PRED, (int)NN);
}